// ForceModel_90400471646700
// MI455X (gfx1250) — compile-verified
//
#include <hip/hip_runtime.h>
#include <math.h>

// ---------------------------------------------------------------------------
// Force-field energy (bonds + angles + impropers + propers), B batches.
// Stage 1: fused per-term energy, batch-major grid, block partial sums -> ws.
// Stage 2: deterministic per-batch reduction; cross-lane sum done with the
//          CDNA5 fp32 WMMA (v_wmma_f32_16x16x4_f32) + xor shuffles.
// ---------------------------------------------------------------------------

typedef float v2f __attribute__((ext_vector_type(2)));
typedef float v8f __attribute__((ext_vector_type(8)));

struct F3 { float x, y, z; };

__device__ __forceinline__ F3 f3sub(F3 a, F3 b) { return {a.x - b.x, a.y - b.y, a.z - b.z}; }
__device__ __forceinline__ float f3dot(F3 a, F3 b) { return fmaf(a.x, b.x, fmaf(a.y, b.y, a.z * b.z)); }
__device__ __forceinline__ F3 f3cross(F3 a, F3 b) {
    return { a.y * b.z - a.z * b.y, a.z * b.x - a.x * b.z, a.x * b.y - a.y * b.x };
}
__device__ __forceinline__ F3 f3scale(F3 a, float s) { return {a.x * s, a.y * s, a.z * s}; }

__device__ __forceinline__ F3 loadPos(const float* __restrict__ posB, int atom) {
    const float* p = posB + (size_t)atom * 3u;   // 12B gather, dword aligned
    F3 r; r.x = p[0]; r.y = p[1]; r.z = p[2]; return r;
}

__device__ __forceinline__ float dihedral(F3 p0, F3 p1, F3 p2, F3 p3) {
    F3 b0 = f3sub(p0, p1);                 // -(p1 - p0)
    F3 b1 = f3sub(p2, p1);
    F3 b2 = f3sub(p3, p2);
    float inv = 1.0f / sqrtf(f3dot(b1, b1));
    b1 = f3scale(b1, inv);
    F3 v = f3sub(b0, f3scale(b1, f3dot(b0, b1)));
    F3 w = f3sub(b2, f3scale(b1, f3dot(b2, b1)));
    float x = f3dot(v, w);
    float y = f3dot(f3cross(b1, v), w);
    return atan2f(y, x);
}

__global__ __launch_bounds__(256) void ff_energy_stage1(
    const float* __restrict__ pos,
    const int*   __restrict__ bond_idcs,     const int*   __restrict__ bond_type,
    const float* __restrict__ equ_bond,      const float* __restrict__ k_bond,
    const int*   __restrict__ angle_idcs,    const int*   __restrict__ angle_type,
    const float* __restrict__ equ_angle,     const float* __restrict__ k_angle,
    const int*   __restrict__ improper_idcs, const int*   __restrict__ improper_type,
    const float* __restrict__ equ_improper,  const float* __restrict__ k_improper,
    const int*   __restrict__ proper_idcs,   const float* __restrict__ proper_phase,
    const float* __restrict__ proper_const,  const float* __restrict__ proper_mul,
    float* __restrict__ partials,
    int nAtoms, int nBond, int nAngle, int nImp, int nProp, int blocksPerBatch)
{
    const int  b    = blockIdx.y;
    const float* posB = pos + (size_t)b * (size_t)nAtoms * 3u;
    const int nAB   = nBond + nAngle;
    const int nABI  = nAB + nImp;
    const int nTerms = nABI + nProp;
    const int stride = blocksPerBatch * 256;

    float acc = 0.0f;
    for (int t = blockIdx.x * 256 + threadIdx.x; t < nTerms; t += stride) {
        if (t < nBond) {
            const int i  = t;
            const int a0 = bond_idcs[2 * i + 0];
            const int a1 = bond_idcs[2 * i + 1];
            F3 dv = f3sub(loadPos(posB, a0), loadPos(posB, a1));
            float d  = sqrtf(f3dot(dv, dv));
            int   ty = bond_type[i];
            float dd = d - equ_bond[ty];
            acc = fmaf(k_bond[ty] * dd, dd, acc);
        } else if (t < nAB) {
            const int i  = t - nBond;
            const int a0 = angle_idcs[3 * i + 0];
            const int a1 = angle_idcs[3 * i + 1];
            const int a2 = angle_idcs[3 * i + 2];
            F3 p1 = loadPos(posB, a1);
            F3 v1 = f3sub(loadPos(posB, a0), p1);
            F3 v2 = f3sub(loadPos(posB, a2), p1);
            float c = f3dot(v1, v2) / sqrtf(f3dot(v1, v1) * f3dot(v2, v2));
            c = fminf(1.0f, fmaxf(-1.0f, c));
            float th = acosf(c);
            int   ty = angle_type[i];
            float dt = th - equ_angle[ty];
            acc = fmaf(k_angle[ty] * dt, dt, acc);
        } else if (t < nABI) {
            const int i  = t - nAB;
            const int a0 = improper_idcs[4 * i + 0];
            const int a1 = improper_idcs[4 * i + 1];
            const int a2 = improper_idcs[4 * i + 2];
            const int a3 = improper_idcs[4 * i + 3];
            float phi = dihedral(loadPos(posB, a0), loadPos(posB, a1),
                                 loadPos(posB, a2), loadPos(posB, a3));
            int   ty = improper_type[i];
            float dp = phi - equ_improper[ty];
            acc = fmaf(k_improper[ty] * dp, dp, acc);
        } else {
            const int i  = t - nABI;
            const int a0 = proper_idcs[4 * i + 0];
            const int a1 = proper_idcs[4 * i + 1];
            const int a2 = proper_idcs[4 * i + 2];
            const int a3 = proper_idcs[4 * i + 3];
            float phi = dihedral(loadPos(posB, a0), loadPos(posB, a1),
                                 loadPos(posB, a2), loadPos(posB, a3));
            acc += proper_const[i] *
                   (1.0f + cosf(fmaf(proper_mul[i], phi, -proper_phase[i])));
        }
    }

    // Intra-wave (wave32) reduction, fixed order -> deterministic.
    #pragma unroll
    for (int m = 16; m >= 1; m >>= 1) acc += __shfl_xor(acc, m, 32);

    __shared__ float lds[8];
    const int lane = threadIdx.x & 31;
    const int wv   = threadIdx.x >> 5;
    if (lane == 0) lds[wv] = acc;
    __syncthreads();
    if (wv == 0) {
        float s = (lane < 8) ? lds[lane] : 0.0f;
        #pragma unroll
        for (int m = 4; m >= 1; m >>= 1) s += __shfl_xor(s, m, 32);
        if (lane == 0)
            partials[(size_t)b * (size_t)blocksPerBatch + blockIdx.x] = s;
    }
}

// Stage 2: one wave per batch. Lane-strided fp32 accumulation (fixed order),
// then a 32->16 cross-lane sum via V_WMMA_F32_16X16X4_F32:
//   A = ones(16x4), B holds acc in column n = lane%16 (one K slot per lane,
//   other slot zero), C = 0  =>  D[m][n] = acc[n] + acc[n+16] exactly (fp32).
// Finish the 16 column sums with 4 xor shuffles.
__global__ __launch_bounds__(32) void ff_energy_stage2(
    const float* __restrict__ partials, float* __restrict__ out, int nPartials)
{
    const int b    = blockIdx.x;
    const int lane = threadIdx.x;
    const float* p = partials + (size_t)b * (size_t)nPartials;

    float acc = 0.0f;
    for (int i = lane; i < nPartials; i += 32) acc += p[i];

    v2f a;  a.x = 1.0f; a.y = 1.0f;          // A = ones (16x4)
    v2f bm; bm.x = acc; bm.y = 0.0f;         // B: acc at (k?, n=lane%16)
    v8f c = {};
    v8f d = __builtin_amdgcn_wmma_f32_16x16x4_f32(
        /*neg_a=*/false, a, /*neg_b=*/false, bm,
        /*c_mod=*/(short)0, c, /*reuse_a=*/false, /*reuse_b=*/false);

    float r = d[0];                          // = acc[n] + acc[n+16], n = lane%16
    #pragma unroll
    for (int m = 8; m >= 1; m >>= 1) r += __shfl_xor(r, m, 32);

    if (lane == 0) out[b] = r;
}

extern "C" void kernel_launch(void* const* d_in, const int* in_sizes, int n_in,
                              void* d_out, int out_size, void* d_ws, size_t ws_size,
                              hipStream_t stream) {
    const float* pos           = (const float*)d_in[0];
    const int*   bond_idcs     = (const int*)  d_in[1];
    const int*   bond_type     = (const int*)  d_in[2];
    const float* equ_bond      = (const float*)d_in[3];
    const float* k_bond        = (const float*)d_in[4];
    const int*   angle_idcs    = (const int*)  d_in[5];
    const int*   angle_type    = (const int*)  d_in[6];
    const float* equ_angle     = (const float*)d_in[7];
    const float* k_angle       = (const float*)d_in[8];
    const int*   improper_idcs = (const int*)  d_in[9];
    const int*   improper_type = (const int*)  d_in[10];
    const float* equ_improper  = (const float*)d_in[11];
    const float* k_improper    = (const float*)d_in[12];
    const int*   proper_idcs   = (const int*)  d_in[13];
    const float* proper_phase  = (const float*)d_in[14];
    const float* proper_const  = (const float*)d_in[15];
    const float* proper_mul    = (const float*)d_in[16];

    const int Bn     = out_size;                       // one energy per batch
    const int nBond  = in_sizes[1] / 2;
    const int nAngle = in_sizes[5] / 3;
    const int nImp   = in_sizes[9] / 4;
    const int nProp  = in_sizes[13] / 4;
    const int nAtoms = in_sizes[0] / (3 * Bn);

    int blocksPerBatch = 512;                          // 512*256 threads/batch
    while (blocksPerBatch > 1 &&
           (size_t)Bn * (size_t)blocksPerBatch * sizeof(float) > ws_size)
        blocksPerBatch >>= 1;

    float* partials = (float*)d_ws;

    dim3 grid1((unsigned)blocksPerBatch, (unsigned)Bn, 1);
    ff_energy_stage1<<<grid1, 256, 0, stream>>>(
        pos, bond_idcs, bond_type, equ_bond, k_bond,
        angle_idcs, angle_type, equ_angle, k_angle,
        improper_idcs, improper_type, equ_improper, k_improper,
        proper_idcs, proper_phase, proper_const, proper_mul,
        partials, nAtoms, nBond, nAngle, nImp, nProp, blocksPerBatch);

    ff_energy_stage2<<<dim3((unsigned)Bn, 1, 1), 32, 0, stream>>>(
        partials, (float*)d_out, blocksPerBatch);
}